// GraphConvolution_16449724743811
// MI455X (gfx1250) — compile-verified
//
#include <hip/hip_runtime.h>

// GraphConvolution for MI455X (gfx1250, wave32).
//   support = x @ W           -> WMMA f32 16x16x4 (memory-bound: keep full f32)
//   out[d] += ew * support[s] -> global_atomic_add_f32 (L2-resident scatter)
//   out = relu(out)
//
// d_ws layout: support[N_NODES * OUT_F] floats (25.6 MB required).

typedef __attribute__((ext_vector_type(2))) float v2f;
typedef __attribute__((ext_vector_type(8))) float v8f;

#define IN_F  256
#define OUT_F 64
// Transposed-W LDS row stride (floats). 260 mod 64 == 4 -> the 16 lanes of a
// half-wave hit banks {4n,4n+1} and the other half {4n+2,4n+3}: all 64 banks,
// zero conflicts for the b64 B-fragment loads.
#define WT_STRIDE 260

// ---------------------------------------------------------------- zero / relu
__global__ __launch_bounds__(256) void zero_f4(float4* __restrict__ p, int n4) {
  int i = blockIdx.x * blockDim.x + threadIdx.x;
  if (i < n4) p[i] = make_float4(0.f, 0.f, 0.f, 0.f);
}

__global__ __launch_bounds__(256) void relu_f4(float4* __restrict__ p, int n4) {
  int i = blockIdx.x * blockDim.x + threadIdx.x;
  if (i < n4) {
    float4 v = p[i];
    v.x = fmaxf(v.x, 0.f); v.y = fmaxf(v.y, 0.f);
    v.z = fmaxf(v.z, 0.f); v.w = fmaxf(v.w, 0.f);
    p[i] = v;
  }
}

// ---------------------------------------------------------------- WMMA GEMM
// Block = 256 threads = 8 waves. Each wave computes a 16-row strip of
// support (16 x 64) = 4 accumulator tiles of 16x16 f32, K = 256 in steps of 4.
//
// f32 16x4 A-matrix layout (ISA 7.12.2): lane l (0..31) holds row M = l&15,
// VGPR0/1 = K = 2*(l>>4) + {0,1}.  B (4x16) mirrors it: lane holds col
// N = l&15, K = 2*(l>>4) + {0,1}.  C/D 16x16 f32: VGPR r, lanes 0-15 ->
// (M=r, N=lane), lanes 16-31 -> (M=r+8, N=lane-16).
//
// W is staged TRANSPOSED in LDS (wldsT[n][k], padded stride) so every B
// fragment is one contiguous 8-byte ds_load_b64 straight into an even-aligned
// VGPR pair -- no v_mov repacking in the inner loop.
__global__ __launch_bounds__(256) void gemm_wmma_f32(
    const float* __restrict__ x, const float* __restrict__ w,
    float* __restrict__ support, int n_nodes) {
  __shared__ float wldsT[OUT_F * WT_STRIDE];  // ~65 KB

  // Cooperative stage of W (row-major [k][n] in memory) into transposed LDS.
  for (int i = threadIdx.x; i < IN_F * OUT_F; i += blockDim.x) {
    int k = i >> 6;        // 0..255
    int n = i & 63;        // 0..63
    wldsT[n * WT_STRIDE + k] = w[i];
  }
  __syncthreads();

  const int wave = threadIdx.x >> 5;
  const int lane = threadIdx.x & 31;
  const int rowTile = blockIdx.x * 8 + wave;        // 16-row strip index
  if (rowTile * 16 >= n_nodes) return;              // wave-uniform exit
  const int row0 = rowTile * 16;

  const int m  = lane & 15;        // A: row within tile / B: col within tile
  const int k2 = (lane >> 4) * 2;  // K sub-offset {0,2}

  const float* __restrict__ xrow = x + (size_t)(row0 + m) * IN_F;

  v8f acc[4] = {v8f{}, v8f{}, v8f{}, v8f{}};

  for (int kb = 0; kb < IN_F; kb += 4) {
    // A fragment: two consecutive f32 -> one global_load_b64
    v2f a = *(const v2f*)(xrow + kb + k2);
#pragma unroll
    for (int nt = 0; nt < 4; ++nt) {
      // B fragment: contiguous in transposed LDS -> one ds_load_b64
      v2f b = *(const v2f*)(&wldsT[(nt * 16 + m) * WT_STRIDE + kb + k2]);
      acc[nt] = __builtin_amdgcn_wmma_f32_16x16x4_f32(
          /*neg_a=*/false, a, /*neg_b=*/false, b,
          /*c_mod=*/(short)0, acc[nt], /*reuse_a=*/false, /*reuse_b=*/false);
    }
  }

  // Store D tiles: lanes 0-15 cover M = r, lanes 16-31 cover M = r + 8.
  const int ncol  = lane & 15;
  const int mbase = (lane >> 4) * 8;
#pragma unroll
  for (int nt = 0; nt < 4; ++nt) {
#pragma unroll
    for (int r = 0; r < 8; ++r) {
      support[(size_t)(row0 + mbase + r) * OUT_F + nt * 16 + ncol] = acc[nt][r];
    }
  }
}

// ---------------------------------------------------------------- edge scatter
// One wave per edge: edge scalars are wave-uniform -> scalar loads; each lane
// reads a b64 of the source row (coalesced 256 B per wave) and issues two
// non-returning global_atomic_add_f32 (no packed-f32 atomic exists on CDNA5).
// support (25.6 MB) and out (25.6 MB) both stay hot in the 192 MB L2.
__global__ __launch_bounds__(256) void spmm_scatter(
    const float* __restrict__ support, const int* __restrict__ esrc,
    const int* __restrict__ edst, const float* __restrict__ ew,
    float* __restrict__ out, int n_edges) {
  long long g = (long long)blockIdx.x * 256 + threadIdx.x;
  int e = (int)(g >> 5);           // wave-uniform edge id
  if (e >= n_edges) return;
  int c = (int)(g & 31) * 2;       // feature pair
  int s = esrc[e];
  int d = edst[e];
  float wv = ew[e];
  v2f sv = *(const v2f*)(support + (size_t)s * OUT_F + c);
  float* op = out + (size_t)d * OUT_F + c;
  atomicAdd(op + 0, sv.x * wv);
  atomicAdd(op + 1, sv.y * wv);
}

// ---------------------------------------------------------------- launcher
extern "C" void kernel_launch(void* const* d_in, const int* in_sizes, int n_in,
                              void* d_out, int out_size, void* d_ws, size_t ws_size,
                              hipStream_t stream) {
  const float* x    = (const float*)d_in[0];
  const float* w    = (const float*)d_in[1];
  const int*   esrc = (const int*)d_in[2];
  const int*   edst = (const int*)d_in[3];
  const float* ew   = (const float*)d_in[4];
  float* out     = (float*)d_out;
  float* support = (float*)d_ws;  // n_nodes * OUT_F floats

  const int n_nodes = in_sizes[0] / IN_F;
  const int n_edges = in_sizes[2];
  const int out_elems = n_nodes * OUT_F;

  // 1) zero the accumulator (harness poisons d_out)
  const int n4 = out_elems / 4;
  zero_f4<<<(n4 + 255) / 256, 256, 0, stream>>>((float4*)out, n4);

  // 2) support = x @ W   (8 waves/block, 16 rows/wave)
  const int rowTiles = (n_nodes + 15) / 16;
  const int gblocks  = (rowTiles + 7) / 8;
  gemm_wmma_f32<<<gblocks, 256, 0, stream>>>(x, w, support, n_nodes);

  // 3) scatter-add messages (1 wave per edge)
  const long long sthreads = (long long)n_edges * 32;
  const int sblocks = (int)((sthreads + 255) / 256);
  spmm_scatter<<<sblocks, 256, 0, stream>>>(support, esrc, edst, ew, out, n_edges);

  // 4) relu in place
  relu_f4<<<(n4 + 255) / 256, 256, 0, stream>>>((float4*)out, n4);
}